// DeepseekV3TopkRouter_88106959110535
// MI455X (gfx1250) — compile-verified
//
#include <hip/hip_runtime.h>
#include <stdint.h>

// DeepseekV3 top-k router for MI455X (gfx1250, wave32).
// One token per wave: lane l owns experts [8l, 8l+8). 1KB of logits per token
// is DMA'd global->LDS with gfx1250 async loads, double buffered per wave.
// Pipeline per iteration: issue DMA(t+1) -> s_wait_asynccnt<=2 (async loads
// complete in order, so <=2 outstanding means buffer t is ready while t+1's
// DMA is in flight) -> compute t. No WMMA: the op has no matmul structure;
// memory floor is ~6us (134MB @ 23.3TB/s) and the work is wave32 reductions.

#define WAVES_PER_BLOCK 8
#define THREADS_PER_BLOCK (WAVES_PER_BLOCK * 32)

__device__ __forceinline__ float sigmoid_fast(float x) {
  // Used identically in scoring and in the final gather -> bit-identical.
  return __builtin_amdgcn_rcpf(1.0f + __expf(-x));
}

__device__ __forceinline__ unsigned long long shflx64(unsigned long long v, int m) {
  int lo = __shfl_xor((int)(unsigned)(v & 0xFFFFFFFFull), m, 32);
  int hi = __shfl_xor((int)(unsigned)(v >> 32), m, 32);
  return ((unsigned long long)(unsigned)hi << 32) | (unsigned)(unsigned int)lo;
}

// Async DMA: copy 32B per lane (global -> LDS). Per CDNA5 ISA,
// the instruction offset is added to BOTH the LDS and global addresses, so one
// (ldsaddr, goff) register pair serves both 16B halves.
__device__ __forceinline__ void async_copy_32B(uint32_t ldsaddr, uint32_t goff,
                                               uint64_t gbase) {
  asm volatile(
      "global_load_async_to_lds_b128 %0, %1, %2\n\t"
      "global_load_async_to_lds_b128 %0, %1, %2 offset:16"
      :
      : "v"(ldsaddr), "v"(goff), "s"(gbase)
      : "memory");
}

__device__ __forceinline__ void wait_async0() {
  asm volatile("s_wait_asynccnt 0x0" ::: "memory");
}
// Allow the 2 just-issued prefetch ops for the NEXT buffer to stay in flight;
// in-order completion guarantees the CURRENT buffer's 2 ops are done.
__device__ __forceinline__ void wait_async2() {
  asm volatile("s_wait_asynccnt 0x2" ::: "memory");
}

#define CSWAP_DESC(i, j)                       \
  {                                            \
    unsigned long long _a = q[i], _b = q[j];   \
    bool _c = _a < _b;                         \
    q[i] = _c ? _b : _a;                       \
    q[j] = _c ? _a : _b;                       \
  }

__global__ __launch_bounds__(THREADS_PER_BLOCK) void
deepseekv3_topk_router_kernel(const float* __restrict__ logits,
                              const float* __restrict__ bias,
                              int* __restrict__ out_idx,
                              float* __restrict__ out_w, int ntok) {
  const int lane = threadIdx.x & 31;
  const int wslot = threadIdx.x >> 5;
  const int waveId = blockIdx.x * WAVES_PER_BLOCK + wslot;
  const int nwaves = gridDim.x * WAVES_PER_BLOCK;

  // Per-wave double buffer: 2 x 256 floats (1KB each).
  __shared__ __align__(16) float smem[WAVES_PER_BLOCK * 2 * 256];

  const uint64_t gbase = (uint64_t)(uintptr_t)logits;

  // Per-lane correction bias for experts [8*lane, 8*lane+8) — loop invariant.
  float bb[8];
  {
    const float4* bp = (const float4*)(bias + lane * 8);
    float4 b0 = bp[0], b1 = bp[1];
    bb[0] = b0.x; bb[1] = b0.y; bb[2] = b0.z; bb[3] = b0.w;
    bb[4] = b1.x; bb[5] = b1.y; bb[6] = b1.z; bb[7] = b1.w;
  }
  // Key low words: 255 - global_expert_index (lower index => larger key).
  unsigned klo[8];
#pragma unroll
  for (int j = 0; j < 8; ++j) klo[j] = 255u - (unsigned)(lane * 8 + j);

  const int g = lane >> 2;  // this lane's expert group (0..7)

  int sel = 0;
  int tok = waveId;
  if (tok < ntok) {
    uint32_t la = (uint32_t)(uintptr_t)(&smem[(wslot * 2 + 0) * 256 + lane * 8]);
    async_copy_32B(la, ((uint32_t)tok << 10) + (uint32_t)(lane * 32), gbase);
  }

  for (; tok < ntok; tok += nwaves) {
    // Prefetch next token's tile first, then wait only for the current one.
    int nxt = tok + nwaves;
    if (nxt < ntok) {
      uint32_t la =
          (uint32_t)(uintptr_t)(&smem[(wslot * 2 + (sel ^ 1)) * 256 + lane * 8]);
      async_copy_32B(la, ((uint32_t)nxt << 10) + (uint32_t)(lane * 32), gbase);
      wait_async2();  // current 2 ops done (in-order); next 2 stay in flight
    } else {
      wait_async0();
    }

    const float* buf = &smem[(wslot * 2 + sel) * 256];

    // ---- scores: biased = sigmoid(logit) + bias -------------------------
    float lv[8];
    {
      const float4* lb = (const float4*)(buf + lane * 8);
      float4 x0 = lb[0], x1 = lb[1];
      lv[0] = x0.x; lv[1] = x0.y; lv[2] = x0.z; lv[3] = x0.w;
      lv[4] = x1.x; lv[5] = x1.y; lv[6] = x1.z; lv[7] = x1.w;
    }
    float bv[8];
#pragma unroll
    for (int j = 0; j < 8; ++j) bv[j] = sigmoid_fast(lv[j]) + bb[j];

    // ---- group score = sum of top-2 biased within the 4-lane cluster ----
    float m1 = -INFINITY, m2 = -INFINITY;
#pragma unroll
    for (int j = 0; j < 8; ++j) {
      float v = bv[j];
      float nm1 = fmaxf(m1, v);
      m2 = fmaxf(fminf(m1, v), m2);
      m1 = nm1;
    }
#pragma unroll
    for (int m = 1; m <= 2; m <<= 1) {
      float o1 = __shfl_xor(m1, m, 32);
      float o2 = __shfl_xor(m2, m, 32);
      float nm1 = fmaxf(m1, o1);
      m2 = fmaxf(fminf(m1, o1), fmaxf(m2, o2));
      m1 = nm1;
    }
    float gs = m1 + m2;  // uniform within each 4-lane cluster

    // ---- top-4 groups via rank counting (exact top_k tie semantics) -----
    float gsa[8];
#pragma unroll
    for (int h = 0; h < 8; ++h) gsa[h] = __shfl(gs, h * 4, 32);
    float myg = gsa[g];
    int cnt = 0;
#pragma unroll
    for (int h = 0; h < 8; ++h)
      cnt += (gsa[h] > myg) || ((gsa[h] == myg) && (h < g));
    bool selgrp = cnt < 4;

    // ---- pack sortable keys: (ordered_float << 32) | (255 - idx) --------
    unsigned long long q[8];
#pragma unroll
    for (int j = 0; j < 8; ++j) {
      float mv = selgrp ? bv[j] : 0.0f;  // reference masks to +0.0
      unsigned u = __float_as_uint(mv);
      u ^= (u & 0x80000000u) ? 0xFFFFFFFFu : 0x80000000u;
      q[j] = ((unsigned long long)u << 32) | (unsigned long long)klo[j];
    }

    // ---- sort 8 keys descending (optimal 19-comparator network) ---------
    CSWAP_DESC(0, 1) CSWAP_DESC(2, 3) CSWAP_DESC(4, 5) CSWAP_DESC(6, 7)
    CSWAP_DESC(0, 2) CSWAP_DESC(1, 3) CSWAP_DESC(4, 6) CSWAP_DESC(5, 7)
    CSWAP_DESC(1, 2) CSWAP_DESC(5, 6) CSWAP_DESC(0, 4) CSWAP_DESC(3, 7)
    CSWAP_DESC(1, 5) CSWAP_DESC(2, 6)
    CSWAP_DESC(1, 4) CSWAP_DESC(3, 6)
    CSWAP_DESC(2, 4) CSWAP_DESC(3, 5)
    CSWAP_DESC(3, 4)

    // ---- extract global top-8: lane heads + 5-step u64 butterfly max ----
    unsigned long long rec = 0ull;
#pragma unroll
    for (int k = 0; k < 8; ++k) {
      unsigned long long w = q[0];
#pragma unroll
      for (int m = 16; m >= 1; m >>= 1) {
        unsigned long long o = shflx64(w, m);
        w = (o > w) ? o : w;
      }
      rec = (lane == k) ? w : rec;  // lane k keeps the k-th winner
      bool hit = (q[0] == w);       // unique owner (indices packed in key)
#pragma unroll
      for (int j = 0; j < 7; ++j) q[j] = hit ? q[j + 1] : q[j];
      q[7] = hit ? 0ull : q[7];
    }

    // ---- weights: gather UNbiased sigmoid at winning indices ------------
    unsigned wi = 255u - (unsigned)(rec & 0xFFull);  // lanes>=8: wi=255, in range
    float ws = sigmoid_fast(buf[wi]);                // bit-identical recompute
    float sum = ws;
#pragma unroll
    for (int m = 4; m >= 1; m >>= 1) sum += __shfl_xor(sum, m, 32);
    float wout = ws / (sum + 1e-20f) * 2.5f;

    if (lane < 8) {
      out_idx[(size_t)tok * 8 + lane] = (int)wi;
      out_w[(size_t)tok * 8 + lane] = wout;
    }

    sel ^= 1;
  }
}

extern "C" void kernel_launch(void* const* d_in, const int* in_sizes, int n_in,
                              void* d_out, int out_size, void* d_ws,
                              size_t ws_size, hipStream_t stream) {
  (void)n_in; (void)d_ws; (void)ws_size; (void)out_size;
  const float* logits = (const float*)d_in[0];
  const float* bias = (const float*)d_in[1];
  int ntok = in_sizes[0] / 256;

  int* oi = (int*)d_out;
  float* ow = (float*)d_out + (size_t)ntok * 8;

  int blocks = 4096;  // 32768 waves -> 4 tokens/wave at N=131072 (grid-stride)
  int max_blocks = (ntok + WAVES_PER_BLOCK - 1) / WAVES_PER_BLOCK;
  if (blocks > max_blocks) blocks = max_blocks;
  if (blocks < 1) blocks = 1;

  deepseekv3_topk_router_kernel<<<blocks, THREADS_PER_BLOCK, 0, stream>>>(
      logits, bias, oi, ow, ntok);
}